// GCN_39633958207589
// MI455X (gfx1250) — compile-verified
//
#include <hip/hip_runtime.h>
#include <hip/hip_bf16.h>
#include <stdint.h>

#define D_FEA     48
#define TILE_E    1024
#define SPMM_BLK  256
#define SPMM_GRID 640
#define EW_BLK    256

#if defined(__has_builtin)
#  if __has_builtin(__builtin_amdgcn_tensor_load_to_lds)
#    define HAVE_TDM 1
#  endif
#endif

// ---------------------------------------------------------------------------
// Fallback path: per-lane async global->LDS copies (ASYNCcnt-tracked).
// Compiles/emits on this toolchain (round 1); only used if TDM builtin absent.
// ---------------------------------------------------------------------------
#if !HAVE_TDM
__device__ __forceinline__ void async_copy_b64(void* lds_dst, const void* g_src) {
  unsigned int loff = (unsigned int)(uintptr_t)lds_dst;
  asm volatile("global_load_async_to_lds_b64 %0, %1, off"
               :: "v"(loff), "v"(g_src)
               : "memory");
}
#endif

// ---------------------------------------------------------------------------
// TDM path: one tensor_load_to_lds per array per tile (1-D tensor tile),
// issued by a single wave, EXEC-ignored, tracked with TENSORcnt.
// Descriptor layout per CDNA5 ISA.
// ---------------------------------------------------------------------------
#if HAVE_TDM
typedef unsigned int v4u __attribute__((ext_vector_type(4)));
typedef int          v4i __attribute__((ext_vector_type(4)));
typedef int          v8i __attribute__((ext_vector_type(8)));

__device__ __forceinline__ void tdm_load_1d(void* lds_dst, const void* g_src,
                                            unsigned int nElemRemain) {
  const unsigned long long ga = (unsigned long long)(uintptr_t)g_src;
  v4u g0; v8i g1;
  // Group 0: count=1 | lds_addr | global_addr[56:0] | type=2
  g0[0] = 1u;                                     // count=1, user descriptor
  g0[1] = (unsigned int)(uintptr_t)lds_dst;       // lds_addr (byte offset)
  g0[2] = (unsigned int)ga;                       // global_addr[31:0]
  g0[3] = (unsigned int)(ga >> 32) | (2u << 30);  // global_addr[56:32] | type=2
  // Group 1: data_size=4B; tensor_dim0 = remaining elems (OOB guard);
  //          tensor_dim1=1; tile_dim0=TILE_E; tile_dim1/2 unused (1-D tile)
  g1[0] = (int)(2u << 16);                                     // data_size=2 (4B)
  g1[1] = (int)((nElemRemain & 0xFFFFu) << 16);                // tensor_dim0[15:0]
  g1[2] = (int)(((nElemRemain >> 16) & 0xFFFFu) | (1u << 16)); // dim0[31:16], dim1=1
  g1[3] = (int)((unsigned)TILE_E << 16);                       // tile_dim0
  g1[4] = 0;                                                   // tile_dim1=0, tile_dim2=0
  g1[5] = (int)TILE_E;                                         // tensor_dim0_stride lo
  g1[6] = 0;
  g1[7] = 0;
#if __clang_major__ >= 23
  __builtin_amdgcn_tensor_load_to_lds(g0, g1, (v4i)0, (v4i)0, (v8i)0, 0);
#else
  __builtin_amdgcn_tensor_load_to_lds(g0, g1, (v4i)0, (v4i)0, 0);
#endif
}
#endif  // HAVE_TDM

__device__ __forceinline__ void stage_wait() {
#if HAVE_TDM
  __builtin_amdgcn_s_wait_tensorcnt((short)0);
#else
  asm volatile("s_wait_asynccnt 0" ::: "memory");
#endif
}

// ---------------------------------------------------------------------------
// Stage one tile (TILE_E edges) of row/col/val into LDS.
// Full tiles: TDM (one wave issues 3 DMA descriptors) or per-lane async copy.
// Tail tile: guarded synchronous loads by all threads.
// ---------------------------------------------------------------------------
__device__ __forceinline__ void stage_tile(int* dR, int* dC, float* dV,
                                           const int* __restrict__ erow,
                                           const int* __restrict__ ecol,
                                           const float* __restrict__ eval,
                                           long long base, int nE, int tid) {
  if (base + TILE_E <= (long long)nE) {
#if HAVE_TDM
    if (tid == 0) {
      const unsigned int remain = (unsigned int)((long long)nE - base);
      tdm_load_1d(dR, erow + base, remain);
      tdm_load_1d(dC, ecol + base, remain);
      tdm_load_1d(dV, eval + base, remain);
    }
#else
    const int idx = tid * (TILE_E / SPMM_BLK);   // 4 elements / 16B per thread
    const long long g = base + idx;
    async_copy_b64(dR + idx,     erow + g);
    async_copy_b64(dR + idx + 2, erow + g + 2);
    async_copy_b64(dC + idx,     ecol + g);
    async_copy_b64(dC + idx + 2, ecol + g + 2);
    async_copy_b64(dV + idx,     eval + g);
    async_copy_b64(dV + idx + 2, eval + g + 2);
#endif
  } else {
    const int idx = tid * (TILE_E / SPMM_BLK);
#pragma unroll
    for (int k = 0; k < TILE_E / SPMM_BLK; ++k) {
      const long long gg = base + idx + k;
      if (gg < (long long)nE) {
        dR[idx + k] = erow[gg];
        dC[idx + k] = ecol[gg];
        dV[idx + k] = eval[gg];
      }
    }
  }
}

// ---------------------------------------------------------------------------
// SpMM scatter: y[row[e], :] += val[e] * x[col[e], :]
// 16 lanes per edge; lane L handles features {L, L+16, L+32} -> coalesced
// 64B gathers (L2-resident x) and coalesced global_atomic_add_f32 scatters.
// Edge metadata streamed through LDS, double-buffered TDM DMA.
// y must be pre-zeroed.
// ---------------------------------------------------------------------------
__global__ void __launch_bounds__(SPMM_BLK)
spmm_scatter(const int* __restrict__ erow, const int* __restrict__ ecol,
             const float* __restrict__ eval, const float* __restrict__ x,
             float* __restrict__ y, int nE) {
  __shared__ int   sRow[2][TILE_E];
  __shared__ int   sCol[2][TILE_E];
  __shared__ float sVal[2][TILE_E];

  const int tid    = threadIdx.x;
  const int lane   = tid & 15;   // feature lane within edge group
  const int grp    = tid >> 4;   // edge subgroup 0..15
  const int nTiles = (nE + TILE_E - 1) / TILE_E;

  int t   = (int)blockIdx.x;
  int buf = 0;

  if (t < nTiles) {
    stage_tile(&sRow[0][0], &sCol[0][0], &sVal[0][0],
               erow, ecol, eval, (long long)t * TILE_E, nE, tid);
  }

  for (; t < nTiles; t += gridDim.x, buf ^= 1) {
    // Issuer's DMA landed; barrier publishes LDS to all waves and guarantees
    // the buffer about to be overwritten was fully consumed.
    stage_wait();
    __syncthreads();

    const int nt = t + gridDim.x;
    if (nt < nTiles) {
      stage_tile(&sRow[buf ^ 1][0], &sCol[buf ^ 1][0], &sVal[buf ^ 1][0],
                 erow, ecol, eval, (long long)nt * TILE_E, nE, tid);
    }

    const long long base = (long long)t * TILE_E;
#pragma unroll 4
    for (int p = 0; p < TILE_E / 16; ++p) {
      const int el = p * 16 + grp;
      const long long e = base + el;
      // Prefetch next pass's gather line while this pass computes.
      const int el2 = el + 16;
      if (el2 < TILE_E && base + el2 < (long long)nE) {
        const int c2 = sCol[buf][el2];
        __builtin_prefetch(x + (long long)c2 * D_FEA + lane, 0, 1);
      }
      if (e < (long long)nE) {
        const int   r = sRow[buf][el];
        const int   c = sCol[buf][el];
        const float v = sVal[buf][el];
        const float* xr = x + (long long)c * D_FEA + lane;
        const float a0 = xr[0];
        const float a1 = xr[16];
        const float a2 = xr[32];
        float* yr = y + (long long)r * D_FEA + lane;
        unsafeAtomicAdd(yr +  0, v * a0);   // global_atomic_add_f32
        unsafeAtomicAdd(yr + 16, v * a1);
        unsafeAtomicAdd(yr + 32, v * a2);
      }
    }
    __syncthreads();
  }
  stage_wait();
}

// ---------------------------------------------------------------------------
// Elementwise kernels (float4 vectorized, B128 traffic)
// ---------------------------------------------------------------------------
__global__ void gcn_init(const float4* __restrict__ fea, float4* __restrict__ lA,
                         float4* __restrict__ lB, float4* __restrict__ acc, int n4) {
  const int i = blockIdx.x * blockDim.x + threadIdx.x;
  if (i < n4) {
    const float4 f = fea[i];
    lA[i]  = f;                                  // learn = fea
    acc[i] = f;                                  // tmp   = fea
    lB[i]  = make_float4(0.f, 0.f, 0.f, 0.f);    // zero first scatter target
  }
}

// dst (spmm result) += bias; acc = (acc + dst) * finalScale;
// optionally re-zero the consumed source buffer for the next layer's scatter.
// finalScale is 1.0f for inner layers (exact) and 0.25f on the last layer,
// folding the reference's "/ (N_GNN+1)" and saving a full output pass.
__global__ void gcn_post(float4* __restrict__ dst, const float* __restrict__ brow,
                         float4* __restrict__ acc, float4* __restrict__ zbuf,
                         float finalScale, int n4) {
  const int i = blockIdx.x * blockDim.x + threadIdx.x;
  if (i >= n4) return;
  const int d = (i * 4) % D_FEA;        // 48 % 4 == 0 -> contiguous, 16B aligned
  const float4 b = *(const float4*)(brow + d);
  float4 v = dst[i];
  v.x += b.x; v.y += b.y; v.z += b.z; v.w += b.w;
  dst[i] = v;
  float4 a = acc[i];
  a.x = (a.x + v.x) * finalScale;
  a.y = (a.y + v.y) * finalScale;
  a.z = (a.z + v.z) * finalScale;
  a.w = (a.w + v.w) * finalScale;
  acc[i] = a;
  if (zbuf) zbuf[i] = make_float4(0.f, 0.f, 0.f, 0.f);
}

// ---------------------------------------------------------------------------
// Host-side launcher
// inputs: fea[N*48] f32, adj_row[E] i32, adj_col[E] i32, adj_val[E] f32,
//         bias[3*48] f32 ; output: [N*48] f32
// ---------------------------------------------------------------------------
extern "C" void kernel_launch(void* const* d_in, const int* in_sizes, int n_in,
                              void* d_out, int out_size, void* d_ws, size_t ws_size,
                              hipStream_t stream) {
  const float* fea  = (const float*)d_in[0];
  const int*   erow = (const int*)d_in[1];
  const int*   ecol = (const int*)d_in[2];
  const float* eval = (const float*)d_in[3];
  const float* bias = (const float*)d_in[4];

  const int ND = in_sizes[0];      // N * 48
  const int E  = in_sizes[1];
  const int n4 = ND / 4;

  float* learnA = (float*)d_ws;          // N*48 floats
  float* learnB = learnA + ND;           // N*48 floats
  float* acc    = (float*)d_out;         // accumulator lives in the output

  const int ewGrid = (n4 + EW_BLK - 1) / EW_BLK;

  gcn_init<<<ewGrid, EW_BLK, 0, stream>>>((const float4*)fea, (float4*)learnA,
                                          (float4*)learnB, (float4*)acc, n4);

  // layer 0: learnA -> learnB
  spmm_scatter<<<SPMM_GRID, SPMM_BLK, 0, stream>>>(erow, ecol, eval, learnA, learnB, E);
  gcn_post<<<ewGrid, EW_BLK, 0, stream>>>((float4*)learnB, bias + 0 * D_FEA,
                                          (float4*)acc, (float4*)learnA, 1.0f, n4);
  // layer 1: learnB -> learnA
  spmm_scatter<<<SPMM_GRID, SPMM_BLK, 0, stream>>>(erow, ecol, eval, learnB, learnA, E);
  gcn_post<<<ewGrid, EW_BLK, 0, stream>>>((float4*)learnA, bias + 1 * D_FEA,
                                          (float4*)acc, (float4*)learnB, 1.0f, n4);
  // layer 2: learnA -> learnB
  spmm_scatter<<<SPMM_GRID, SPMM_BLK, 0, stream>>>(erow, ecol, eval, learnA, learnB, E);
  gcn_post<<<ewGrid, EW_BLK, 0, stream>>>((float4*)learnB, bias + 2 * D_FEA,
                                          (float4*)acc, (float4*)nullptr, 0.25f, n4);
}